// DARNN_7834020348399
// MI455X (gfx1250) — compile-verified
//
#include <hip/hip_runtime.h>
#include <math.h>

// ---------------------------------------------------------------------------
// DA-RNN for MI455X (gfx1250, wave32).  All matmuls use V_WMMA_F32_16X16X4_F32
// (fp32 in/out: the 767-step recurrent chain is precision-sensitive and the
// kernel is L2/transcendental bound, so low-precision WMMA buys nothing).
// Working set (~140 MB) fits the 192 MB L2, so per-step re-reads of the
// hoisted attention precomputes (XW, EW) never touch HBM.
// ---------------------------------------------------------------------------

constexpr int kB = 128;   // batch
constexpr int kT = 256;   // time steps
constexpr int kN = 256;   // input features
constexpr int kM = 256;   // encoder / top-LSTM units
constexpr int kP = 256;   // decoder units
constexpr int k4M = 1024; // 4*M (= 4*P)

typedef float v2f __attribute__((ext_vector_type(2)));
typedef float v8f __attribute__((ext_vector_type(8)));

__device__ __forceinline__ v8f wmma_f32(v2f a, v2f b, v8f c) {
  // D = A(16x4) x B(4x16) + C(16x16), fp32
  return __builtin_amdgcn_wmma_f32_16x16x4_f32(false, a, false, b, (short)0, c,
                                               false, false);
}

// ---------------------------------------------------------------------------
// Generic fp32 WMMA GEMM: C(Mr x Nc) = A(Mr x K) * Bw(K x Nc) + bias(Nc)
// One wave = one 16x16 output tile.  Mr,Nc multiples of 16; K multiple of 4.
// Wave-uniform early-out keeps EXEC all-1s for every WMMA.
// ---------------------------------------------------------------------------
__global__ void gemm16(const float* __restrict__ A, const float* __restrict__ Bw,
                       const float* __restrict__ bias, float* __restrict__ C,
                       int Mr, int Nc, int K, int lda, int ldb, int ldc) {
  const int wavesPerBlk = blockDim.x >> 5;
  int wave = blockIdx.x * wavesPerBlk + (threadIdx.x >> 5);
  int tilesN = Nc >> 4;
  int totalTiles = (Mr >> 4) * tilesN;
  if (wave >= totalTiles) return;                  // uniform per wave
  int tm = (wave / tilesN) << 4;
  int tn = (wave % tilesN) << 4;
  int lane = threadIdx.x & 31;
  int half = lane >> 4;                            // 0: K pair {0,1}, 1: {2,3}
  int l16  = lane & 15;

  v8f acc = {0.f, 0.f, 0.f, 0.f, 0.f, 0.f, 0.f, 0.f};
  const float* Ap = A + (size_t)(tm + l16) * lda + 2 * half;  // A[M=l16][k+2h]
  const float* Bp = Bw + (size_t)(2 * half) * ldb + tn + l16; // B[k+2h][N=l16]
  for (int k = 0; k < K; k += 4) {
    v2f a, b;
    a[0] = Ap[k];
    a[1] = Ap[k + 1];
    b[0] = Bp[(size_t)k * ldb];
    b[1] = Bp[(size_t)(k + 1) * ldb];
    acc = wmma_f32(a, b, acc);
  }
  int col = tn + l16;
  float badd = bias ? bias[col] : 0.f;
#pragma unroll
  for (int r = 0; r < 8; ++r) {                    // VGPR r: M=r (+8 for hi half)
    int row = tm + r + 8 * half;
    C[(size_t)row * ldc + col] = acc[r] + badd;
  }
}

// ---------------------------------------------------------------------------
// XW[b,n,t] = sum_tt enc[b,tt,n] * W1i[tt,t] + b1i[t]
// A is read "transposed" out of enc_data (stride N along K).  Rows = B*N.
// ---------------------------------------------------------------------------
__global__ void xw_gemm(const float* __restrict__ enc, const float* __restrict__ W1i,
                        const float* __restrict__ b1i, float* __restrict__ XW) {
  const int wavesPerBlk = blockDim.x >> 5;
  int wave = blockIdx.x * wavesPerBlk + (threadIdx.x >> 5);
  int tilesN = kT >> 4;
  int totalTiles = ((kB * kN) >> 4) * tilesN;
  if (wave >= totalTiles) return;
  int tm = (wave / tilesN) << 4;
  int tn = (wave % tilesN) << 4;
  int lane = threadIdx.x & 31, half = lane >> 4, l16 = lane & 15;

  int row = tm + l16;          // row = b*N + n  (tiles never straddle b: N%16==0)
  int b   = row >> 8;          // / kN
  int n   = row & (kN - 1);
  const float* Ap = enc + (size_t)b * kT * kN + (size_t)(2 * half) * kN + n;
  const float* Bp = W1i + (size_t)(2 * half) * kT + tn + l16;

  v8f acc = {0.f, 0.f, 0.f, 0.f, 0.f, 0.f, 0.f, 0.f};
  for (int k = 0; k < kT; k += 4) {
    v2f a, bb;
    a[0] = Ap[(size_t)k * kN];
    a[1] = Ap[(size_t)(k + 1) * kN];
    bb[0] = Bp[(size_t)k * kT];
    bb[1] = Bp[(size_t)(k + 1) * kT];
    acc = wmma_f32(a, bb, acc);
  }
  int col = tn + l16;
  float badd = b1i[col];
#pragma unroll
  for (int r = 0; r < 8; ++r) {
    int orow = tm + r + 8 * half;
    XW[(size_t)orow * kT + col] = acc[r] + badd;
  }
}

// ---------------------------------------------------------------------------
// LSTM pointwise: z (B,4U) gates i,f,g,o ; updates h,c; optional q=[h,c] and
// optional h-sequence store at step t.
// ---------------------------------------------------------------------------
__global__ void lstm_pointwise(const float* __restrict__ z, float* __restrict__ h,
                               float* __restrict__ c, float* __restrict__ q,
                               float* __restrict__ hseq, int t, int units) {
  int idx = blockIdx.x * blockDim.x + threadIdx.x;
  if (idx >= kB * units) return;
  int b = idx / units, j = idx - b * units;
  const float* zb = z + (size_t)b * 4 * units;
  float iv = zb[j];
  float fv = zb[units + j];
  float gv = zb[2 * units + j];
  float ov = zb[3 * units + j];
  float si = 1.f / (1.f + __expf(-iv));
  float sf = 1.f / (1.f + __expf(-fv));
  float so = 1.f / (1.f + __expf(-ov));
  float c2 = sf * c[idx] + si * tanhf(gv);
  float h2 = so * tanhf(c2);
  c[idx] = c2;
  h[idx] = h2;
  if (q) {
    q[(size_t)b * 2 * units + j] = h2;
    q[(size_t)b * 2 * units + units + j] = c2;
  }
  if (hseq) hseq[((size_t)b * kT + t) * units + j] = h2;
}

// xh[b] = [ X[b,t,:] (N) | h[b,:] (M) ]
__global__ void concat_xh(const float* __restrict__ X, const float* __restrict__ h,
                          float* __restrict__ xh, int t) {
  int idx = blockIdx.x * blockDim.x + threadIdx.x;
  if (idx >= kB * (kN + kM)) return;
  int b = idx / (kN + kM), j = idx - b * (kN + kM);
  xh[idx] = (j < kN) ? X[((size_t)b * kT + t) * kN + j]
                     : h[(size_t)b * kM + (j - kN)];
}

// enc_out = enc_data * alpha, in place on the alpha buffer.  The reference's
// scrambled reshape of the (T,B,1,N) alpha stack is exactly a linear
// reinterpretation of this flat buffer, so a linear elementwise multiply is
// bit-faithful.
__global__ void apply_alpha(const float* __restrict__ enc, float* __restrict__ a,
                            size_t nTot) {
  size_t idx = (size_t)blockIdx.x * blockDim.x + threadIdx.x;
  if (idx >= nTot) return;
  a[idx] = enc[idx] * a[idx];
}

// Encoder input attention for step t.  One block per batch b, 256 threads
// (thread = n).  s[n] = sum_t tanh(XW[b,n,t] + qW[b,t]) * vi[t] + bvi,
// softmax over n, stored at alpha[(t*B+b)*N + n].
__global__ void enc_attn(const float* __restrict__ XW, const float* __restrict__ qW,
                         const float* __restrict__ vi, const float* __restrict__ bvi,
                         float* __restrict__ alpha, int t) {
  int b = blockIdx.x;
  int n = threadIdx.x;
  __shared__ float qs[kT];
  __shared__ float vs[kT];
  __shared__ float red[kN];
  qs[n] = qW[(size_t)b * kT + n];
  vs[n] = vi[n];
  __syncthreads();
  const float* xwr = XW + ((size_t)b * kN + n) * kT;
  float s = bvi[0];
  for (int tt = 0; tt < kT; ++tt) s += tanhf(xwr[tt] + qs[tt]) * vs[tt];
  // softmax over n
  red[n] = s;
  __syncthreads();
  for (int st = kN / 2; st > 0; st >>= 1) {
    if (n < st) red[n] = fmaxf(red[n], red[n + st]);
    __syncthreads();
  }
  float mx = red[0];
  __syncthreads();
  float e = __expf(s - mx);
  red[n] = e;
  __syncthreads();
  for (int st = kN / 2; st > 0; st >>= 1) {
    if (n < st) red[n] += red[n + st];
    __syncthreads();
  }
  alpha[((size_t)t * kB + b) * kN + n] = e / red[0];
}

// Decoder temporal attention: one block per b, 256 threads.
// sc[t] = sum_m tanh(EW[b,t,m] + qW[b,m]) * vt[m] + bvt, beta = softmax over t,
// ctx[b,m] = sum_t beta[t] * enc_h[b,t,m].
__global__ void temporal_attn(const float* __restrict__ EW, const float* __restrict__ qW,
                              const float* __restrict__ vt, const float* __restrict__ bvt,
                              const float* __restrict__ enc_h, float* __restrict__ ctx) {
  int b = blockIdx.x;
  int j = threadIdx.x;           // first a t index, then an m index
  __shared__ float qs[kM];
  __shared__ float vs[kM];
  __shared__ float red[kT];
  __shared__ float beta[kT];
  qs[j] = qW[(size_t)b * kM + j];
  vs[j] = vt[j];
  __syncthreads();
  const float* er = EW + ((size_t)b * kT + j) * kM;
  float s = bvt[0];
  for (int m = 0; m < kM; ++m) s += tanhf(er[m] + qs[m]) * vs[m];
  red[j] = s;
  __syncthreads();
  for (int st = kT / 2; st > 0; st >>= 1) {
    if (j < st) red[j] = fmaxf(red[j], red[j + st]);
    __syncthreads();
  }
  float mx = red[0];
  __syncthreads();
  float e = __expf(s - mx);
  red[j] = e;
  __syncthreads();
  for (int st = kT / 2; st > 0; st >>= 1) {
    if (j < st) red[j] += red[j + st];
    __syncthreads();
  }
  beta[j] = e / red[0];
  __syncthreads();
  // ctx over m = j
  float acc = 0.f;
  const float* hb = enc_h + (size_t)b * kT * kM + j;
  for (int tt = 0; tt < kT; ++tt) acc += beta[tt] * hb[(size_t)tt * kM];
  ctx[(size_t)b * kM + j] = acc;
}

// Decoder input: xi = [x_t, ctx] @ Wd + bd (scalar per b); then build padded
// concat row xh[b] = [xi, h (P), 0,0,0] of width 260 (K multiple of 4).
__global__ void dec_input(const float* __restrict__ dec_data, const float* __restrict__ ctx,
                          const float* __restrict__ h, const float* __restrict__ Wd,
                          const float* __restrict__ bd, float* __restrict__ xh, int t) {
  int b = blockIdx.x, j = threadIdx.x;   // 256 threads
  __shared__ float red[kM];
  red[j] = Wd[1 + j] * ctx[(size_t)b * kM + j];
  __syncthreads();
  for (int st = kM / 2; st > 0; st >>= 1) {
    if (j < st) red[j] += red[j + st];
    __syncthreads();
  }
  float* xb = xh + (size_t)b * 260;
  if (j == 0) xb[0] = red[0] + Wd[0] * dec_data[(size_t)b * (kT - 1) + t] + bd[0];
  xb[1 + j] = h[(size_t)b * kP + j];
  if (j < 3) xb[257 + j] = 0.f;
}

// Stack W (256,1024) over U (256,1024) -> (512,1024)
__global__ void build_cat2(const float* __restrict__ W, const float* __restrict__ U,
                           float* __restrict__ out) {
  int idx = blockIdx.x * blockDim.x + threadIdx.x;
  if (idx >= 512 * k4M) return;
  int r = idx / k4M, col = idx - r * k4M;
  out[idx] = (r < 256) ? W[(size_t)r * k4M + col] : U[(size_t)(r - 256) * k4M + col];
}

// [Wdec (1,1024); Udec (256,1024); zeros (3,1024)] -> (260,1024)
__global__ void build_catD(const float* __restrict__ Wdec, const float* __restrict__ Udec,
                           float* __restrict__ out) {
  int idx = blockIdx.x * blockDim.x + threadIdx.x;
  if (idx >= 260 * k4M) return;
  int r = idx / k4M, col = idx - r * k4M;
  float v = 0.f;
  if (r == 0) v = Wdec[col];
  else if (r <= 256) v = Udec[(size_t)(r - 1) * k4M + col];
  out[idx] = v;
}

__global__ void fill0(float* __restrict__ p, int nElem) {
  int i = blockIdx.x * blockDim.x + threadIdx.x;
  if (i < nElem) p[i] = 0.f;
}

// xh[b] = [h (P) | ctx (M)]
__global__ void concat_hctx(const float* __restrict__ h, const float* __restrict__ ctx,
                            float* __restrict__ xh) {
  int idx = blockIdx.x * blockDim.x + threadIdx.x;
  if (idx >= kB * (kP + kM)) return;
  int b = idx / (kP + kM), j = idx - b * (kP + kM);
  xh[idx] = (j < kP) ? h[(size_t)b * kP + j] : ctx[(size_t)b * kM + (j - kP)];
}

// out[b] = dec1[b,:] . W2o + b2o
__global__ void final_out(const float* __restrict__ dec1, const float* __restrict__ W2o,
                          const float* __restrict__ b2o, float* __restrict__ out) {
  int b = blockIdx.x, j = threadIdx.x;   // 256 threads
  __shared__ float red[kP];
  red[j] = dec1[(size_t)b * kP + j] * W2o[j];
  __syncthreads();
  for (int st = kP / 2; st > 0; st >>= 1) {
    if (j < st) red[j] += red[j + st];
    __syncthreads();
  }
  if (j == 0) out[b] = red[0] + b2o[0];
}

// ---------------------------------------------------------------------------
extern "C" void kernel_launch(void* const* d_in, const int* in_sizes, int n_in,
                              void* d_out, int out_size, void* d_ws, size_t ws_size,
                              hipStream_t stream) {
  (void)in_sizes; (void)n_in; (void)out_size;
  const float* enc_data = (const float*)d_in[0];
  const float* dec_data = (const float*)d_in[1];
  const float* We   = (const float*)d_in[2];
  const float* Ue   = (const float*)d_in[3];
  const float* be   = (const float*)d_in[4];
  const float* W1i  = (const float*)d_in[5];
  const float* b1i  = (const float*)d_in[6];
  const float* W2i  = (const float*)d_in[7];
  const float* b2i  = (const float*)d_in[8];
  const float* vi   = (const float*)d_in[9];
  const float* bvi  = (const float*)d_in[10];
  const float* Wl   = (const float*)d_in[11];
  const float* Ul   = (const float*)d_in[12];
  const float* bl   = (const float*)d_in[13];
  const float* W1t  = (const float*)d_in[14];
  const float* b1t  = (const float*)d_in[15];
  const float* W2t  = (const float*)d_in[16];
  const float* b2t  = (const float*)d_in[17];
  const float* vt   = (const float*)d_in[18];
  const float* bvt  = (const float*)d_in[19];
  const float* Wd   = (const float*)d_in[20];
  const float* bd   = (const float*)d_in[21];
  const float* Wdec = (const float*)d_in[22];
  const float* Udec = (const float*)d_in[23];
  const float* bdec = (const float*)d_in[24];
  const float* W1o  = (const float*)d_in[25];
  const float* b1o  = (const float*)d_in[26];
  const float* W2o  = (const float*)d_in[27];
  const float* b2o  = (const float*)d_in[28];
  float* out = (float*)d_out;

  // ---- workspace layout (fp32) ----
  float* ws = (float*)d_ws;
  size_t off = 0;
  auto alloc = [&](size_t n) { float* p = ws + off; off += n; return p; };
  float* XW    = alloc((size_t)kB * kN * kT);   // 33.5 MB
  float* EW    = alloc((size_t)kB * kT * kM);   // 33.5 MB
  float* enc_h = alloc((size_t)kB * kT * kM);   // 33.5 MB
  float* alpha = alloc((size_t)kB * kT * kN);   // 33.5 MB (becomes enc_out in place)
  float* WcatE = alloc((size_t)512 * k4M);
  float* WcatL = alloc((size_t)512 * k4M);
  float* WcatD = alloc((size_t)260 * k4M);
  float* xh    = alloc((size_t)kB * 512);       // also used at width 260
  float* z     = alloc((size_t)kB * k4M);
  float* h     = alloc((size_t)kB * 256);
  float* c     = alloc((size_t)kB * 256);
  float* q     = alloc((size_t)kB * 512);
  float* qW    = alloc((size_t)kB * 256);
  float* ctx   = alloc((size_t)kB * 256);
  float* dec1  = alloc((size_t)kB * 256);
  if (ws_size < off * sizeof(float)) return;    // scratch too small: bail safely

  auto gemm = [&](const float* A, const float* Bw, const float* bias, float* C,
                  int Mr, int Nc, int K, int lda, int ldb, int ldc) {
    int tiles = (Mr / 16) * (Nc / 16);
    int blocks = (tiles + 3) / 4;                // 4 waves (128 thr) per block
    gemm16<<<blocks, 128, 0, stream>>>(A, Bw, bias, C, Mr, Nc, K, lda, ldb, ldc);
  };

  // ---- one-time precomputes ----
  build_cat2<<<(512 * k4M) / 256, 256, 0, stream>>>(We, Ue, WcatE);
  build_cat2<<<(512 * k4M) / 256, 256, 0, stream>>>(Wl, Ul, WcatL);
  build_catD<<<(260 * k4M) / 256, 256, 0, stream>>>(Wdec, Udec, WcatD);
  {
    int tiles = ((kB * kN) / 16) * (kT / 16);
    xw_gemm<<<(tiles + 3) / 4, 128, 0, stream>>>(enc_data, W1i, b1i, XW);
  }

  // ---- encoder with input attention ----
  fill0<<<(kB * kM) / 256, 256, 0, stream>>>(h, kB * kM);
  fill0<<<(kB * kM) / 256, 256, 0, stream>>>(c, kB * kM);
  for (int t = 0; t < kT; ++t) {
    concat_xh<<<(kB * 512) / 256, 256, 0, stream>>>(enc_data, h, xh, t);
    gemm(xh, WcatE, be, z, kB, k4M, 512, 512, k4M, k4M);
    lstm_pointwise<<<(kB * kM) / 256, 256, 0, stream>>>(z, h, c, q, nullptr, t, kM);
    gemm(q, W2i, b2i, qW, kB, kT, 512, 512, kT, kT);
    enc_attn<<<kB, 256, 0, stream>>>(XW, qW, vi, bvi, alpha, t);
  }
  {
    size_t nTot = (size_t)kB * kT * kN;
    apply_alpha<<<(unsigned)((nTot + 255) / 256), 256, 0, stream>>>(enc_data, alpha, nTot);
  }

  // ---- top-level LSTM (return_sequences -> enc_h) ----
  fill0<<<(kB * kM) / 256, 256, 0, stream>>>(h, kB * kM);
  fill0<<<(kB * kM) / 256, 256, 0, stream>>>(c, kB * kM);
  for (int t = 0; t < kT; ++t) {
    concat_xh<<<(kB * 512) / 256, 256, 0, stream>>>(alpha /*enc_out*/, h, xh, t);
    gemm(xh, WcatL, bl, z, kB, k4M, 512, 512, k4M, k4M);
    lstm_pointwise<<<(kB * kM) / 256, 256, 0, stream>>>(z, h, c, nullptr, enc_h, t, kM);
  }

  // EW = enc_h @ W1t + b1t  (hoisted out of the decoder loop)
  gemm(enc_h, W1t, b1t, EW, kB * kT, kM, kM, kM, kM, kM);

  // ---- decoder with temporal attention ----
  fill0<<<(kB * kP) / 256, 256, 0, stream>>>(h, kB * kP);
  fill0<<<(kB * kP) / 256, 256, 0, stream>>>(c, kB * kP);
  fill0<<<(kB * kM) / 256, 256, 0, stream>>>(ctx, kB * kM);
  for (int t = 0; t < kT - 1; ++t) {
    dec_input<<<kB, 256, 0, stream>>>(dec_data, ctx, h, Wd, bd, xh, t);
    gemm(xh, WcatD, bdec, z, kB, k4M, 260, 260, k4M, k4M);
    lstm_pointwise<<<(kB * kP) / 256, 256, 0, stream>>>(z, h, c, q, nullptr, t, kP);
    gemm(q, W2t, b2t, qW, kB, kM, 512, 512, kM, kM);
    temporal_attn<<<kB, 256, 0, stream>>>(EW, qW, vt, bvt, enc_h, ctx);
  }

  // ---- output head ----
  concat_hctx<<<(kB * 512) / 256, 256, 0, stream>>>(h, ctx, xh);
  gemm(xh, W1o, b1o, dec1, kB, kP, 512, 512, kP, kP);
  final_out<<<kB, 256, 0, stream>>>(dec1, W2o, b2o, out);
}